// GNNLayer_56341380989568
// MI455X (gfx1250) — compile-verified
//
#include <hip/hip_runtime.h>

typedef __attribute__((ext_vector_type(2))) float v2f;
typedef __attribute__((ext_vector_type(8))) float v8f;

#define FDIM 48   // IN_F == HID_F == OUT_F == 48

// ---------------- degree ----------------
__global__ void deg_kernel(const int* __restrict__ dst, float* __restrict__ deg, int nEdges) {
    int e = blockIdx.x * blockDim.x + threadIdx.x;
    if (e < nEdges) atomicAdd(&deg[dst[e]], 1.0f);
}

__global__ void invdeg_kernel(float* __restrict__ deg, int n) {
    int v = blockIdx.x * blockDim.x + threadIdx.x;
    if (v < n) deg[v] = 1.0f / fmaxf(deg[v], 1.0f);
}

// ------- scatter-add x[src[e]] -> agg[dst[e]], float4 granularity (48 = 12*4) -------
__global__ void scatter_kernel(const float* __restrict__ x, const int* __restrict__ src,
                               const int* __restrict__ dst, float* __restrict__ agg,
                               int nEdges) {
    unsigned tid = blockIdx.x * blockDim.x + threadIdx.x;
    unsigned total = (unsigned)nEdges * 12u;
    if (tid >= total) return;
    unsigned e = tid / 12u;
    unsigned c = tid - e * 12u;
    const float4 v = *(const float4*)(x + (size_t)src[e] * FDIM + c * 4);
    float* a = agg + (size_t)dst[e] * FDIM + c * 4;
    atomicAdd(a + 0, v.x);
    atomicAdd(a + 1, v.y);
    atomicAdd(a + 2, v.z);
    atomicAdd(a + 3, v.w);
}

// ------- fused SAGE layer GEMM: out = x@Wself + (agg*invd)@Wneigh + b, optional ReLU -------
// Persistent wave with VGPR-resident weights:
//   * all 72 B fragments (2 matrices x 12 k-steps x 3 n-tiles, v2f each = 144 VGPRs)
//     are loaded ONCE per wave, then reused across `tilesPerWave` 16-node tiles.
//   * per tile: only A fragments stream in (b64 loads), 72 x v_wmma_f32_16x16x4_f32.
//   * 6 independent accumulator chains (cs[3], cn[3]) to pipeline WMMA latency.
// Exact tiling (N = 6250*16, waves*tilesPerWave == 6250) -> EXEC all-ones everywhere.
__global__ void __launch_bounds__(64)
sage_gemm_kernel(const float* __restrict__ x,      // [N,48] self features
                 const float* __restrict__ agg,    // [N,48] summed neighbor feats
                 const float* __restrict__ invd,   // [N]    1/max(deg,1)
                 const float* __restrict__ Wself,  // [48,48] row-major (k,n)
                 const float* __restrict__ Wneigh, // [48,48]
                 const float* __restrict__ bias,   // [48]
                 float* __restrict__ out,          // [N,48]
                 int relu, int tilesPerWave) {
    const int lane = threadIdx.x & 31;
    const int wave = threadIdx.x >> 5;
    const int half = lane >> 4;   // 0: lanes 0-15, 1: lanes 16-31
    const int l15  = lane & 15;
    const int gw   = blockIdx.x * 2 + wave;          // global wave id

    // ---- load all weight fragments once (VGPR-resident) ----
    v2f wS[12][3], wN[12][3];
    #pragma unroll
    for (int k = 0; k < 12; ++k) {
        const int kk = k * 4 + half * 2;   // this half-wave's K pair: kk, kk+1
        #pragma unroll
        for (int nt = 0; nt < 3; ++nt) {
            const int col = nt * 16 + l15;
            wS[k][nt].x = Wself [(size_t)(kk)     * FDIM + col];
            wS[k][nt].y = Wself [(size_t)(kk + 1) * FDIM + col];
            wN[k][nt].x = Wneigh[(size_t)(kk)     * FDIM + col];
            wN[k][nt].y = Wneigh[(size_t)(kk + 1) * FDIM + col];
        }
    }
    float bv0 = bias[0 * 16 + l15];
    float bv1 = bias[1 * 16 + l15];
    float bv2 = bias[2 * 16 + l15];

    #pragma unroll 1
    for (int t = 0; t < tilesPerWave; ++t) {
        const int m0 = (gw * tilesPerWave + t) * 16;   // base node of this tile
        const float* xrow = x   + (size_t)(m0 + l15) * FDIM;
        const float* grow = agg + (size_t)(m0 + l15) * FDIM;
        const float  dsc  = invd[m0 + l15];

        v8f cs[3], cn[3];
        #pragma unroll
        for (int r = 0; r < 8; ++r) {
            cs[0][r] = bv0; cs[1][r] = bv1; cs[2][r] = bv2;
            cn[0][r] = 0.0f; cn[1][r] = 0.0f; cn[2][r] = 0.0f;
        }

        #pragma unroll
        for (int k = 0; k < 12; ++k) {
            const int kk = k * 4 + half * 2;
            v2f a;
            a.x = xrow[kk];
            a.y = xrow[kk + 1];
            v2f a2;
            a2.x = grow[kk] * dsc;
            a2.y = grow[kk + 1] * dsc;
            #pragma unroll
            for (int nt = 0; nt < 3; ++nt) {
                cs[nt] = __builtin_amdgcn_wmma_f32_16x16x4_f32(
                    false, a, false, wS[k][nt], (short)0, cs[nt], false, false);
                cn[nt] = __builtin_amdgcn_wmma_f32_16x16x4_f32(
                    false, a2, false, wN[k][nt], (short)0, cn[nt], false, false);
            }
        }

        // store 16x48 tile (C layout: VGPR r -> row m0 + half*8 + r, col nt*16 + l15)
        #pragma unroll
        for (int nt = 0; nt < 3; ++nt) {
            #pragma unroll
            for (int r = 0; r < 8; ++r) {
                float v = cs[nt][r] + cn[nt][r];
                if (relu) v = fmaxf(v, 0.0f);
                out[(size_t)(m0 + half * 8 + r) * FDIM + nt * 16 + l15] = v;
            }
        }
    }
}

extern "C" void kernel_launch(void* const* d_in, const int* in_sizes, int n_in,
                              void* d_out, int out_size, void* d_ws, size_t ws_size,
                              hipStream_t stream) {
    const float* features = (const float*)d_in[0];
    const int*   src      = (const int*)d_in[1];
    const int*   dst      = (const int*)d_in[2];
    const float* Wself1   = (const float*)d_in[3];
    const float* Wneigh1  = (const float*)d_in[4];
    const float* b1       = (const float*)d_in[5];
    const float* Wself2   = (const float*)d_in[6];
    const float* Wneigh2  = (const float*)d_in[7];
    const float* b2       = (const float*)d_in[8];
    float* out = (float*)d_out;

    const int N = in_sizes[0] / FDIM;   // 100000
    const int E = in_sizes[1];          // 1600000

    // workspace layout: invd[N] | agg[N*48] | h[N*48]
    char* ws = (char*)d_ws;
    size_t off = 0;
    float* invd = (float*)(ws + off); off += (((size_t)N * 4)        + 255) & ~(size_t)255;
    float* agg  = (float*)(ws + off); off += (((size_t)N * FDIM * 4) + 255) & ~(size_t)255;
    float* h    = (float*)(ws + off);

    const unsigned scatterTotal  = (unsigned)E * 12u;
    const unsigned scatterBlocks = (scatterTotal + 255u) / 256u;

    // GEMM persistent-wave partition: tiles = N/16 = 6250 = waves * tilesPerWave
    const int tiles        = N / 16;       // 6250
    const int tilesPerWave = 5;
    const int gemmWaves    = tiles / tilesPerWave;   // 1250
    const int gemmBlocks   = gemmWaves / 2;          // 625 (2 waves of 32 per block)

    // degrees (shared by both layers)
    hipMemsetAsync(invd, 0, (size_t)N * 4, stream);
    deg_kernel<<<(E + 255) / 256, 256, 0, stream>>>(dst, invd, E);
    invdeg_kernel<<<(N + 255) / 256, 256, 0, stream>>>(invd, N);

    // ---- layer 1 ----
    hipMemsetAsync(agg, 0, (size_t)N * FDIM * 4, stream);
    scatter_kernel<<<scatterBlocks, 256, 0, stream>>>(features, src, dst, agg, E);
    sage_gemm_kernel<<<gemmBlocks, 64, 0, stream>>>(features, agg, invd,
                                                    Wself1, Wneigh1, b1, h, 1, tilesPerWave);

    // ---- layer 2 ----
    hipMemsetAsync(agg, 0, (size_t)N * FDIM * 4, stream);
    scatter_kernel<<<scatterBlocks, 256, 0, stream>>>(h, src, dst, agg, E);
    sage_gemm_kernel<<<gemmBlocks, 64, 0, stream>>>(h, agg, invd,
                                                    Wself2, Wneigh2, b2, out, 0, tilesPerWave);
}